// PhysicalLayerModule_6313601925848
// MI455X (gfx1250) — compile-verified
//
#include <hip/hip_runtime.h>
#include <math.h>

typedef __attribute__((ext_vector_type(2))) float v2f;
typedef __attribute__((ext_vector_type(8))) float v8f;
typedef __attribute__((vector_size(16))) int b128_t;   // pointee type for async b128 copy

#define NR 64
#define NT 64
#define NRF 8
#define HS 68      // LDS stride for Hr/Hi rows (bank-conflict padding)
#define BS 132     // LDS stride for B^T rows
#define GS 17      // LDS stride for HVA/HQA rows

#define AS1 __attribute__((address_space(1)))
#define AS3 __attribute__((address_space(3)))

struct __align__(16) SM {
    float Hr[NR * HS];        // H real, row-major [r][t]
    float Hi[NR * HS];        // H imag
    float BvT[16 * BS];       // B^T for GEMM1: [n][k], k=0..127 = [VA-part | (+/-)VA-part]
    float BqT[16 * BS];       // B^T for GEMM2 (QA-based)
    float HVA[NR * GS];       // [r][0..7]=Re(HVA), [r][8..15]=Im(HVA)
    float HQA[NT * GS];       // [t][0..7]=Re(HhQA), [t][8..15]=Im(HhQA)
    float sr[NR], si[NR];
    float Xr[NT], Xi[NT];
    float Yr[NR], Yi[NR];
    float Avr[64], Avi[64];   // 8x8 complex
    float Aqr[64], Aqi[64];
    float rvr[8], rvi[8];     // rhs / solution z
    float ur[8], ui[8];       // rhs / solution w
};

// Parallel 8x8 complex Gaussian elimination (no pivoting: A = Gram + sigma2*I is HPD).
// On return br/bi hold the solution. Must be called by ALL threads (uniform).
__device__ __forceinline__ void solve8(float* Ar, float* Ai, float* br, float* bi, int tid)
{
    for (int k = 0; k < 8; ++k) {
        __syncthreads();
        if (tid > k && tid < 8) {
            float dr = Ar[k * 8 + k], di = Ai[k * 8 + k];
            float inv = 1.0f / (dr * dr + di * di);
            float nrm = Ar[tid * 8 + k], nim = Ai[tid * 8 + k];
            float fr = (nrm * dr + nim * di) * inv;
            float fi = (nim * dr - nrm * di) * inv;
            for (int j = k; j < 8; ++j) {
                float pr = Ar[k * 8 + j], pi = Ai[k * 8 + j];
                Ar[tid * 8 + j] -= fr * pr - fi * pi;
                Ai[tid * 8 + j] -= fr * pi + fi * pr;
            }
            float pr = br[k], pi = bi[k];
            br[tid] -= fr * pr - fi * pi;
            bi[tid] -= fr * pi + fi * pr;
        }
    }
    __syncthreads();
    if (tid == 0) {
        for (int k = 7; k >= 0; --k) {
            float sr = br[k], si = bi[k];
            for (int j = k + 1; j < 8; ++j) {
                float pr = Ar[k * 8 + j], pi = Ai[k * 8 + j];
                sr -= pr * br[j] - pi * bi[j];
                si -= pr * bi[j] + pi * br[j];
            }
            float dr = Ar[k * 8 + k], di = Ai[k * 8 + k];
            float inv = 1.0f / (dr * dr + di * di);
            br[k] = (sr * dr + si * di) * inv;
            bi[k] = (si * dr - sr * di) * inv;
        }
    }
    __syncthreads();
}

__global__ __launch_bounds__(128) void phylayer_kernel(
    const float* __restrict__ grid_real, const float* __restrict__ grid_imag,
    const float* __restrict__ phases_t, const float* __restrict__ phases_r,
    const float* __restrict__ H_real, const float* __restrict__ H_imag,
    const float* __restrict__ noise_real, const float* __restrict__ noise_imag,
    float* __restrict__ out)
{
    __shared__ SM sm;
    const int tid = threadIdx.x;
    const int bk = blockIdx.x;
    const size_t hbase = (size_t)bk * (NR * NT);
    const size_t vbase = (size_t)bk * NR;

    // ---- Phase A: async DMA of H into LDS (global -> LDS, no VGPR round-trip).
    // Issued first so the copy overlaps the sincos/B-matrix build below.
    {
        const float* srcR = H_real + hbase;
        const float* srcI = H_imag + hbase;
        for (int q = tid; q < (NR * NT) / 4; q += 128) {
            int r = q >> 4;            // 16 float4 per 64-wide row
            int c = (q & 15) << 2;
            __builtin_amdgcn_global_load_async_to_lds_b128(
                (AS1 b128_t*)(srcR + q * 4), (AS3 b128_t*)&sm.Hr[r * HS + c], 0, 0);
            __builtin_amdgcn_global_load_async_to_lds_b128(
                (AS1 b128_t*)(srcI + q * 4), (AS3 b128_t*)&sm.Hi[r * HS + c], 0, 0);
        }
    }

    // ---- Phase B: build packed B^T matrices from phases; load s (overlaps DMA) ----
    // GEMM1: HVA = H @ VA.  A1 = [Hr | Hi] (K=128).
    //   Re: B[k<64][f]=VAr, B[k>=64][f]=-VAi ; Im: B[k<64][8+f]=VAi, B[k>=64][8+f]=VAr
    // GEMM2: HhQA = H^H @ QA.  A2[t][k] = k<64 ? Hr[k][t] : Hi[k-64][t].
    //   Re: B[k<64][f]=QAr, B[k>=64][f]=QAi ; Im: B[k<64][8+f]=QAi, B[k>=64][8+f]=-QAr
    for (int e = tid; e < NT * NRF; e += 128) {
        int t = e >> 3, f = e & 7;
        float sn, cs;
        sincosf(phases_t[e], &sn, &cs);
        sm.BvT[f * BS + t]            = cs;
        sm.BvT[f * BS + 64 + t]       = -sn;
        sm.BvT[(8 + f) * BS + t]      = sn;
        sm.BvT[(8 + f) * BS + 64 + t] = cs;
        sincosf(phases_r[e], &sn, &cs);
        sm.BqT[f * BS + t]            = cs;   // QAr[r=t][f]
        sm.BqT[f * BS + 64 + t]       = sn;   // QAi
        sm.BqT[(8 + f) * BS + t]      = sn;
        sm.BqT[(8 + f) * BS + 64 + t] = -cs;
    }
    if (tid < NR) {
        sm.sr[tid] = grid_real[vbase + tid];
        sm.si[tid] = grid_imag[vbase + tid];
    }

    // Drain this wave's async copies, then make LDS visible to all waves.
    asm volatile("s_wait_asynccnt 0x0" ::: "memory");
    __syncthreads();

    // ---- Phase C: WMMA GEMMs (fp32, 16x16x4). 4 waves, one M-tile each. ----
    {
        const int lane = tid & 31;
        const int m0 = (tid >> 5) * 16;
        const int m = m0 + (lane & 15);   // A: M = lane%16
        const int n = lane & 15;          // B: N = lane%16 (assumed mirror of A)
        const int kh = lane >> 4;         // per-lane K sub-pair selector

        v8f acc1 = {};  // HVA combined [Re | Im] in N
        v8f acc2 = {};  // HhQA combined

        // GEMM1: A rows of Hr then Hi (K = 0..63 then 64..127)
#pragma unroll
        for (int kk = 0; kk < 16; ++kk) {
            int kb = 4 * kk + 2 * kh;
            v2f a = *reinterpret_cast<const v2f*>(&sm.Hr[m * HS + kb]);
            v2f b = *reinterpret_cast<const v2f*>(&sm.BvT[n * BS + kb]);
            acc1 = __builtin_amdgcn_wmma_f32_16x16x4_f32(false, a, false, b,
                                                         (short)0, acc1, false, false);
        }
#pragma unroll
        for (int kk = 0; kk < 16; ++kk) {
            int kb = 4 * kk + 2 * kh;
            v2f a = *reinterpret_cast<const v2f*>(&sm.Hi[m * HS + kb]);
            v2f b = *reinterpret_cast<const v2f*>(&sm.BvT[n * BS + 64 + kb]);
            acc1 = __builtin_amdgcn_wmma_f32_16x16x4_f32(false, a, false, b,
                                                         (short)0, acc1, false, false);
        }

        // GEMM2: A = H^T access (m is t, k is r)
#pragma unroll
        for (int kk = 0; kk < 16; ++kk) {
            int kb = 4 * kk + 2 * kh;
            v2f a;
            a.x = sm.Hr[kb * HS + m];
            a.y = sm.Hr[(kb + 1) * HS + m];
            v2f b = *reinterpret_cast<const v2f*>(&sm.BqT[n * BS + kb]);
            acc2 = __builtin_amdgcn_wmma_f32_16x16x4_f32(false, a, false, b,
                                                         (short)0, acc2, false, false);
        }
#pragma unroll
        for (int kk = 0; kk < 16; ++kk) {
            int kb = 4 * kk + 2 * kh;
            v2f a;
            a.x = sm.Hi[kb * HS + m];
            a.y = sm.Hi[(kb + 1) * HS + m];
            v2f b = *reinterpret_cast<const v2f*>(&sm.BqT[n * BS + 64 + kb]);
            acc2 = __builtin_amdgcn_wmma_f32_16x16x4_f32(false, a, false, b,
                                                         (short)0, acc2, false, false);
        }

        // D layout: VGPR j -> M = j (lanes 0-15) or j+8 (lanes 16-31), N = lane%16
        int rowbase = m0 + ((lane >> 4) << 3);
        int col = lane & 15;
#pragma unroll
        for (int j = 0; j < 8; ++j) {
            sm.HVA[(rowbase + j) * GS + col] = acc1[j];
            sm.HQA[(rowbase + j) * GS + col] = acc2[j];
        }
    }
    __syncthreads();

    // ---- Phase D: Gram matrices A_v, A_q (+ sigma2*I) and rv = HVA^H s ----
    {
        int f = (tid >> 3) & 7, g = tid & 7;
        if (tid < 64) {
            float accr = (f == g) ? 0.1f : 0.0f, acci = 0.0f;
            for (int r = 0; r < NR; ++r) {
                float ar = sm.HVA[r * GS + f], ai = sm.HVA[r * GS + 8 + f];
                float br = sm.HVA[r * GS + g], bi = sm.HVA[r * GS + 8 + g];
                accr += ar * br + ai * bi;
                acci += ar * bi - ai * br;
            }
            sm.Avr[f * 8 + g] = accr;
            sm.Avi[f * 8 + g] = acci;
        } else {
            float accr = (f == g) ? 0.1f : 0.0f, acci = 0.0f;
            for (int t = 0; t < NT; ++t) {
                float ar = sm.HQA[t * GS + f], ai = sm.HQA[t * GS + 8 + f];
                float br = sm.HQA[t * GS + g], bi = sm.HQA[t * GS + 8 + g];
                accr += ar * br + ai * bi;
                acci += ar * bi - ai * br;
            }
            sm.Aqr[f * 8 + g] = accr;
            sm.Aqi[f * 8 + g] = acci;
        }
        if (tid < 8) {
            float rr = 0.0f, ri = 0.0f;
            for (int r = 0; r < NR; ++r) {
                float ar = sm.HVA[r * GS + tid], ai = sm.HVA[r * GS + 8 + tid];
                rr += ar * sm.sr[r] + ai * sm.si[r];
                ri += ar * sm.si[r] - ai * sm.sr[r];
            }
            sm.rvr[tid] = rr;
            sm.rvi[tid] = ri;
        }
    }

    // z = A_v^{-1} rv  (in-place into rvr/rvi)
    solve8(sm.Avr, sm.Avi, sm.rvr, sm.rvi, tid);

    // ---- Phase F: X = VA z ----
    if (tid < NT) {
        int t = tid;
        float xr = 0.0f, xi = 0.0f;
        for (int f = 0; f < NRF; ++f) {
            float var = sm.BvT[f * BS + t];        // VAr[t][f]
            float vai = sm.BvT[(8 + f) * BS + t];  // VAi[t][f]
            float zr = sm.rvr[f], zi = sm.rvi[f];
            xr += var * zr - vai * zi;
            xi += var * zi + vai * zr;
        }
        sm.Xr[t] = xr;
        sm.Xi[t] = xi;
    }
    __syncthreads();

    // ---- Phase G: Y = H X + sqrt(sigma2/2)*noise ----
    if (tid < NR) {
        int r = tid;
        float yr = 0.0f, yi = 0.0f;
        for (int t = 0; t < NT; ++t) {
            float hr = sm.Hr[r * HS + t], hi = sm.Hi[r * HS + t];
            float xr = sm.Xr[t], xi = sm.Xi[t];
            yr += hr * xr - hi * xi;
            yi += hr * xi + hi * xr;
        }
        const float nsc = 0.22360679774997896f;  // sqrt(0.1/2)
        yr += nsc * noise_real[vbase + r];
        yi += nsc * noise_imag[vbase + r];
        sm.Yr[r] = yr;
        sm.Yi[r] = yi;
    }
    __syncthreads();

    // ---- Phase H: u = QA^H Y ----
    if (tid < 8) {
        float rr = 0.0f, ri = 0.0f;
        for (int r = 0; r < NR; ++r) {
            float qr = sm.BqT[tid * BS + r];       // QAr[r][tid]
            float qi = sm.BqT[tid * BS + 64 + r];  // QAi[r][tid]
            rr += qr * sm.Yr[r] + qi * sm.Yi[r];
            ri += qr * sm.Yi[r] - qi * sm.Yr[r];
        }
        sm.ur[tid] = rr;
        sm.ui[tid] = ri;
    }
    // w = A_q^{-1} u  (solve8 starts with a barrier, covers the u writes)
    solve8(sm.Aqr, sm.Aqi, sm.ur, sm.ui, tid);

    // ---- Phase I: s_hat = HhQA w ; write [real, imag] ----
    if (tid < NT) {
        int t = tid;
        float shr = 0.0f, shi = 0.0f;
        for (int f = 0; f < NRF; ++f) {
            float hr = sm.HQA[t * GS + f], hi = sm.HQA[t * GS + 8 + f];
            float wr = sm.ur[f], wi = sm.ui[f];
            shr += hr * wr - hi * wi;
            shi += hr * wi + hi * wr;
        }
        out[(vbase + t) * 2 + 0] = shr;
        out[(vbase + t) * 2 + 1] = shi;
    }
}

extern "C" void kernel_launch(void* const* d_in, const int* in_sizes, int n_in,
                              void* d_out, int out_size, void* d_ws, size_t ws_size,
                              hipStream_t stream) {
    const float* grid_real  = (const float*)d_in[0];
    const float* grid_imag  = (const float*)d_in[1];
    const float* phases_t   = (const float*)d_in[2];
    const float* phases_r   = (const float*)d_in[3];
    const float* H_real     = (const float*)d_in[4];
    const float* H_imag     = (const float*)d_in[5];
    const float* noise_real = (const float*)d_in[6];
    const float* noise_imag = (const float*)d_in[7];
    float* out = (float*)d_out;

    int BK = in_sizes[4] / (NR * NT);  // 32*512 = 16384
    phylayer_kernel<<<dim3(BK), dim3(128), 0, stream>>>(
        grid_real, grid_imag, phases_t, phases_r,
        H_real, H_imag, noise_real, noise_imag, out);
}